// LatticeQuantizer_53128745452065
// MI455X (gfx1250) — compile-verified
//
#include <hip/hip_runtime.h>
#include <cmath>

// ---------------------------------------------------------------------------
// Hierarchical nested E8 lattice quantizer for gfx1250 (MI455X).
//  - point-per-lane layout for all nonlinear E8 logic (pure VALU, no shuffles)
//  - 8x8 matmuls done on the matrix pipe with V_WMMA_F32_16X16X4_F32
//    (two K=4 chunks chained; 16-point tiles; 2 tiles per wave32)
//  - wave-private LDS slice for point-layout <-> WMMA-layout transposes,
//    all DS addresses precomputed once
//  - power-of-two moduli implemented exactly with trunc / int parity
//  - cround via copysign (bit-identical outputs, 1 v_bfi instead of cmp+sel)
// ---------------------------------------------------------------------------

typedef __attribute__((ext_vector_type(2))) float v2f;
typedef __attribute__((ext_vector_type(8))) float v8f;

#define QF 4.0f
#define NLAYERS 2
#define ROWSTRIDE 12   // floats per LDS row: 48B -> 16B aligned, bank-spread

__device__ __forceinline__ void compiler_fence() {
    __builtin_amdgcn_wave_barrier();
    asm volatile("" ::: "memory");
}

// Exact fmod(v, 4.0f) for f32 (truncated remainder, bit-identical to fmodf):
// v*0.25 exact, trunc exact, 4*t exact, final subtract exact (Sterbenz).
__device__ __forceinline__ float fmod4(float v) {
    return v - 4.0f * truncf(v * 0.25f);
}

// Parity test on an integer-valued float: fmod(s,2)==0  <=>  ((int)s) even.
__device__ __forceinline__ bool is_even(float s) {
    return (((int)s) & 1) == 0;
}

// custom_round: floor(x - sign(x)*eps + 0.5).
// copysign(eps,x) == sign(x)*eps for all x != 0; at x == +/-0 both variants
// still floor to 0, so outputs are bit-identical while lowering to v_bfi_b32.
__device__ __forceinline__ float cround(float x) {
    const float tiny = 1.1920929e-7f; // FLT_EPSILON, matches jnp.finfo(f32).eps
    float y = x - __builtin_copysignf(tiny, x);
    return floorf(y + 0.5f);
}

// g_x with precomputed f = cround(x) and df = x - f (shared with caller):
// flip the coord with max |rounding error| toward/away from x.
__device__ __forceinline__ void g_x8_pre(const float x[8], const float f[8],
                                         const float df[8], float out[8]) {
    float dmax = -1.0f;
    int k = 0;
    #pragma unroll
    for (int i = 0; i < 8; ++i) {
        float d = fabsf(df[i]);                // folds into |src| cmp modifier
        if (d > dmax) { dmax = d; k = i; }     // strict '>' => first max (jnp.argmax)
    }
    // extract x[k], f[k] without dynamic indexing (keep in VGPRs)
    float xk = 0.0f, fk = 0.0f;
    #pragma unroll
    for (int i = 0; i < 8; ++i) {
        bool sel = (i == k);
        xk = sel ? x[i] : xk;
        fk = sel ? f[i] : fk;
    }
    bool go_up = (xk >= 0.0f) ? (fk < xk) : (fk <= xk);
    float newv = go_up ? (fk + 1.0f) : (fk - 1.0f);
    #pragma unroll
    for (int i = 0; i < 8; ++i)
        out[i] = (i == k) ? newv : f[i];
}

// closest point in E8 = D8 union (D8 + 1/2)
__device__ __forceinline__ void cp_e8(const float x[8], float out[8]) {
    // --- integer branch
    float f[8], df0[8], gx[8], y0[8];
    float s0 = 0.0f;
    #pragma unroll
    for (int i = 0; i < 8; ++i) { f[i] = cround(x[i]); s0 += f[i]; }
    #pragma unroll
    for (int i = 0; i < 8; ++i) df0[i] = x[i] - f[i];
    g_x8_pre(x, f, df0, gx);
    bool even0 = is_even(s0);
    #pragma unroll
    for (int i = 0; i < 8; ++i) y0[i] = even0 ? f[i] : gx[i];

    // --- half-integer branch
    float xs[8], fs[8], df1[8], gs[8], y1[8];
    float s1 = 0.0f;
    #pragma unroll
    for (int i = 0; i < 8; ++i) { xs[i] = x[i] - 0.5f; fs[i] = cround(xs[i]); s1 += fs[i]; }
    #pragma unroll
    for (int i = 0; i < 8; ++i) df1[i] = xs[i] - fs[i];
    g_x8_pre(xs, fs, df1, gs);
    bool even1 = is_even(s1);
    #pragma unroll
    for (int i = 0; i < 8; ++i) y1[i] = (even1 ? fs[i] : gs[i]) + 0.5f;

    // --- pick nearer candidate (same elementwise-then-sum order as reference)
    float d0 = 0.0f, d1 = 0.0f;
    #pragma unroll
    for (int i = 0; i < 8; ++i) {
        float a = x[i] - y0[i]; d0 += a * a;
        float b = x[i] - y1[i]; d1 += b * b;
    }
    #pragma unroll
    for (int i = 0; i < 8; ++i) out[i] = (d0 < d1) ? y0[i] : y1[i];
}

// Precomputed per-lane LDS indices (float units) for the transpose staging.
struct MmAddrs {
    int st;       // lane's own row base (store + readback)
    int a0, a1;   // A-gather base, tiles 0/1 (pair at [a], [a+1]; hi chunk +4)
    int d0, d1;   // D-scatter base, tiles 0/1 (element stride ROWSTRIDE)
    bool dvalid;  // this lane scatters a valid D column
};

// Wave-cooperative [32 x 8] * [8 x 8] via two 16-row WMMA tiles, K=8 as 2x K=4.
__device__ __forceinline__ void wave_matmul8x8(float* sl, const MmAddrs& ad,
                                               const float xv[8],
                                               v2f B0, v2f B1, float yv[8]) {
    // stage this lane's point row into LDS
    #pragma unroll
    for (int i = 0; i < 8; ++i) sl[ad.st + i] = xv[i];
    compiler_fence();

    // gather WMMA A layout: lane(row,half), VGPR v -> K = 2*half + v
    v2f a0lo = { sl[ad.a0 + 0], sl[ad.a0 + 1] };
    v2f a0hi = { sl[ad.a0 + 4], sl[ad.a0 + 5] };
    v2f a1lo = { sl[ad.a1 + 0], sl[ad.a1 + 1] };
    v2f a1hi = { sl[ad.a1 + 4], sl[ad.a1 + 5] };

    v8f c0 = {};
    v8f c1 = {};
    // D = A(16x4) * B(4x16) + C, chained over the two K-chunks (full f32 FMA)
    c0 = __builtin_amdgcn_wmma_f32_16x16x4_f32(false, a0lo, false, B0, (short)0, c0, false, false);
    c0 = __builtin_amdgcn_wmma_f32_16x16x4_f32(false, a0hi, false, B1, (short)0, c0, false, false);
    c1 = __builtin_amdgcn_wmma_f32_16x16x4_f32(false, a1lo, false, B0, (short)0, c1, false, false);
    c1 = __builtin_amdgcn_wmma_f32_16x16x4_f32(false, a1hi, false, B1, (short)0, c1, false, false);

    compiler_fence();
    // scatter D (lane n holds column n; VGPR j holds row j (+8 for upper lanes))
    if (ad.dvalid) {
        #pragma unroll
        for (int j = 0; j < 8; ++j) {
            sl[ad.d0 + j * ROWSTRIDE] = c0[j];
            sl[ad.d1 + j * ROWSTRIDE] = c1[j];
        }
    }
    compiler_fence();
    #pragma unroll
    for (int i = 0; i < 8; ++i) yv[i] = sl[ad.st + i];
    compiler_fence();
}

__global__ void __launch_bounds__(256)
LatticeQuantizer_53128745452065_kernel(const float* __restrict__ x,
                                       const float* __restrict__ beta_p,
                                       const float* __restrict__ G,
                                       const float* __restrict__ Ginv,
                                       const float* __restrict__ eps,
                                       float* __restrict__ xhat,
                                       int* __restrict__ codes,
                                       int N) {
    __shared__ float lds[8][32 * ROWSTRIDE];   // 8 wave32 slices, 12 KB total
    const int tid  = blockIdx.x * blockDim.x + threadIdx.x;
    const int lane = threadIdx.x & 31;
    const int wv   = threadIdx.x >> 5;
    float* sl = lds[wv];

    const int idx = (tid < N) ? tid : (N - 1);   // keep wave fully active (EXEC all-1 for WMMA)
    const bool live = (tid < N);
    const float beta = beta_p[0];

    const int row  = lane & 15;
    const int half = lane >> 4;

    // --- precompute all LDS transpose indices once (float units)
    MmAddrs ad;
    ad.st     = lane * ROWSTRIDE;
    ad.a0     = row * ROWSTRIDE + 2 * half;
    ad.a1     = ad.a0 + 16 * ROWSTRIDE;
    ad.d0     = (half * 8) * ROWSTRIDE + row;        // row == colD for lanes with col<8
    ad.d1     = ad.d0 + 16 * ROWSTRIDE;
    ad.dvalid = (row < 8);

    // --- pre-stripe B matrices (8x8 zero-padded to 4x16 chunks) into WMMA B layout
    v2f Gi0, Gi1, Gm0, Gm1;
    {
        #pragma unroll
        for (int c = 0; c < 2; ++c) {
            #pragma unroll
            for (int v = 0; v < 2; ++v) {
                const int k = 4 * c + 2 * half + v;          // K row for this lane/VGPR
                float bi = (row < 8) ? Ginv[k * 8 + row] : 0.0f;
                float bg = (row < 8) ? G[k * 8 + row]    : 0.0f;
                if (c == 0) { Gi0[v] = bi; Gm0[v] = bg; }
                else        { Gi1[v] = bi; Gm1[v] = bg; }
            }
        }
    }
    float e[8];
    #pragma unroll
    for (int i = 0; i < 8; ++i) e[i] = eps[i];

    // --- load this lane's point (two coalesced B128 loads)
    float xl[8];
    {
        const float4* xp = reinterpret_cast<const float4*>(x + (size_t)idx * 8);
        float4 lo = xp[0], hi = xp[1];
        xl[0] = lo.x / beta; xl[1] = lo.y / beta; xl[2] = lo.z / beta; xl[3] = lo.w / beta;
        xl[4] = hi.x / beta; xl[5] = hi.y / beta; xl[6] = hi.z / beta; xl[7] = hi.w / beta;
    }

    // --- encode: M hierarchical levels
    float codes_f[NLAYERS][8];
    #pragma unroll
    for (int layer = 0; layer < NLAYERS; ++layer) {
        float t[8], q[8];
        #pragma unroll
        for (int i = 0; i < 8; ++i) t[i] = xl[i] + e[i];
        cp_e8(t, q);                                     // x_l = Q_E8(x_l + eps)
        float bm[8];
        wave_matmul8x8(sl, ad, q, Gi0, Gi1, bm);         // x_l @ G_inv (WMMA)
        #pragma unroll
        for (int i = 0; i < 8; ++i) codes_f[layer][i] = cround(fmod4(bm[i]));
        #pragma unroll
        for (int i = 0; i < 8; ++i) xl[i] = q[i] * 0.25f;  // /Q exact
    }

    // --- decode: x_hat = beta * sum_i Q^i * (b_i G - Q * Q_E8(b_i G / Q))
    float acc[8];
    #pragma unroll
    for (int i = 0; i < 8; ++i) acc[i] = 0.0f;
    float scale = 1.0f;
    #pragma unroll
    for (int layer = 0; layer < NLAYERS; ++layer) {
        float y[8];
        wave_matmul8x8(sl, ad, codes_f[layer], Gm0, Gm1, y); // b_f @ G (WMMA)
        float yq[8], cp[8];
        #pragma unroll
        for (int i = 0; i < 8; ++i) yq[i] = y[i] * 0.25f;
        cp_e8(yq, cp);
        #pragma unroll
        for (int i = 0; i < 8; ++i) acc[i] += scale * (y[i] - QF * cp[i]);
        scale *= QF;
    }

    // --- store x_hat (f32) and codes (i32), coalesced 128b stores
    if (live) {
        float4* op = reinterpret_cast<float4*>(xhat + (size_t)idx * 8);
        float4 lo, hi;
        lo.x = beta * acc[0]; lo.y = beta * acc[1]; lo.z = beta * acc[2]; lo.w = beta * acc[3];
        hi.x = beta * acc[4]; hi.y = beta * acc[5]; hi.z = beta * acc[6]; hi.w = beta * acc[7];
        op[0] = lo; op[1] = hi;
        #pragma unroll
        for (int layer = 0; layer < NLAYERS; ++layer) {
            int4* cpz = reinterpret_cast<int4*>(codes + (size_t)layer * N * 8 + (size_t)idx * 8);
            int4 a, b;
            a.x = (int)codes_f[layer][0]; a.y = (int)codes_f[layer][1];
            a.z = (int)codes_f[layer][2]; a.w = (int)codes_f[layer][3];
            b.x = (int)codes_f[layer][4]; b.y = (int)codes_f[layer][5];
            b.z = (int)codes_f[layer][6]; b.w = (int)codes_f[layer][7];
            cpz[0] = a; cpz[1] = b;
        }
    }
}

extern "C" void kernel_launch(void* const* d_in, const int* in_sizes, int n_in,
                              void* d_out, int out_size, void* d_ws, size_t ws_size,
                              hipStream_t stream) {
    (void)n_in; (void)out_size; (void)d_ws; (void)ws_size;
    const float* x    = (const float*)d_in[0];
    const float* beta = (const float*)d_in[1];
    const float* G    = (const float*)d_in[2];
    const float* Ginv = (const float*)d_in[3];
    const float* eps  = (const float*)d_in[4];

    const int N = in_sizes[0] / 8;                  // 2097152 points
    float* xhat = (float*)d_out;                    // [N,8] f32
    int*   codes = (int*)d_out + (size_t)N * 8;     // [2,N,8] i32, concatenated after x_hat

    const int block = 256;                           // 8 wave32 per block
    const int grid  = (N + block - 1) / block;
    LatticeQuantizer_53128745452065_kernel<<<grid, block, 0, stream>>>(
        x, beta, G, Ginv, eps, xhat, codes, N);
}